// TemporalPropagator_18356690223185
// MI455X (gfx1250) — compile-verified
//
#include <hip/hip_runtime.h>
#include <hip/hip_bf16.h>
#include <stdint.h>

// ---------------------------------------------------------------------------
// TemporalPropagator for MI455X (gfx1250).
// x_tilde = x @ ((1-w)*U + w*DFT) as ONE fused complex (32768x256)@(256x256)
// matmul on V_WMMA_F32_16X16X4_F32, fused with the h-state recurrence.
// X tiles staged global->LDS by the Tensor Data Mover (TENSOR_LOAD_TO_LDS)
// with hardware padding for a bank-conflict-free pitch.
// ---------------------------------------------------------------------------

#define DIMD     256
#define BATCH    8
#define SPATIAL  4096
#define TBROWS   (BATCH * SPATIAL)   // 32768
#define TILE_M   32                  // rows per block (2 WMMA row-tiles)
#define XPITCH   260                 // LDS row pitch (floats): 256 + 4 pad

typedef __attribute__((ext_vector_type(2))) float    v2f;
typedef __attribute__((ext_vector_type(8))) float    v8f;
typedef __attribute__((ext_vector_type(4))) uint32_t u32x4;
typedef __attribute__((ext_vector_type(4))) int32_t  i32x4;
typedef __attribute__((ext_vector_type(8))) int32_t  i32x8;

#if __has_builtin(__builtin_amdgcn_tensor_load_to_lds) && \
    __has_builtin(__builtin_amdgcn_s_wait_tensorcnt)
#define USE_TDM 1
#endif

// ---------------- workspace layout (float offsets) -------------------------
#define WS_MT_RE   0                         // [256*256] M^T real
#define WS_MT_IM   (WS_MT_RE + DIMD*DIMD)    // [256*256] M^T imag
#define WS_SCAT    (WS_MT_IM + DIMD*DIMD)    // [8*512]   s_cat
#define WS_SRC_RE  (WS_SCAT + BATCH*2*DIMD)  // [8*256]
#define WS_SRC_IM  (WS_SRC_RE + BATCH*DIMD)  // [8*256]
#define WS_GATE    (WS_SRC_IM + BATCH*DIMD)  // [8*256]
#define WS_OPD_RE  (WS_GATE + BATCH*DIMD)    // [256]
#define WS_OPD_IM  (WS_OPD_RE + DIMD)
#define WS_OPF_RE  (WS_OPD_IM + DIMD)
#define WS_OPF_IM  (WS_OPF_RE + DIMD)

// ---------------------------------------------------------------------------
// Build M^T where M = (1-w)*U + w*DFT,  DFT[k][n] = exp(-2pi i k n/256)/16.
// Stored transposed: mt[n*256 + k] so WMMA B-fragments are contiguous in K.
// ---------------------------------------------------------------------------
__global__ void build_m_kernel(const float* __restrict__ U_re,
                               const float* __restrict__ U_im,
                               const float* __restrict__ dft_w,
                               float* __restrict__ mt_re,
                               float* __restrict__ mt_im) {
    int i = blockIdx.x * 256 + threadIdx.x;       // i = k*256 + n
    int k = i >> 8;
    int n = i & 255;
    float w  = 1.0f / (1.0f + __expf(-dft_w[0]));
    int   r  = (k * n) & 255;                     // phase is periodic mod 256
    float th = -6.2831853071795864f * (float)r * (1.0f / 256.0f);
    float c  = __cosf(th);
    float s  = __sinf(th);
    float iw = 1.0f - w;
    float sc = w * (1.0f / 16.0f);                // 1/sqrt(256)
    mt_re[n * DIMD + k] = iw * U_re[i] + sc * c;
    mt_im[n * DIMD + k] = iw * U_im[i] + sc * s;
}

// ---------------------------------------------------------------------------
// Flux path stage 1: x_in = xg_cat @ Wm^T + bm ; flux_next = flux*fdecay + upd
// ---------------------------------------------------------------------------
__global__ void flux_kernel(const float* __restrict__ xg_re,
                            const float* __restrict__ xg_im,
                            const float* __restrict__ flux_re,
                            const float* __restrict__ flux_im,
                            const float* __restrict__ Wm,
                            const float* __restrict__ bm,
                            const float* __restrict__ decay_re,
                            const float* __restrict__ decay_im,
                            float* __restrict__ s_cat,
                            float* __restrict__ out_flux) {
    int idx = blockIdx.x * 256 + threadIdx.x;     // [0, 8*256)
    int b = idx >> 8;
    int n = idx & 255;
    const float* xgr = xg_re + b * DIMD;
    const float* xgi = xg_im + b * DIMD;
    const float* wmR = Wm + (size_t)n * (2 * DIMD);
    const float* wmI = Wm + (size_t)(n + DIMD) * (2 * DIMD);
    float accR = bm[n];
    float accI = bm[n + DIMD];
    for (int k = 0; k < DIMD; ++k) {
        float xr = xgr[k], xi = xgi[k];
        accR = fmaf(xr, wmR[k], fmaf(xi, wmR[k + DIMD], accR));
        accI = fmaf(xr, wmI[k], fmaf(xi, wmI[k + DIMD], accI));
    }
    float fdr = 1.0f / (1.0f + __expf(-decay_re[n]));
    float fdi = decay_im[n];
    float fr = flux_re[idx], fi = flux_im[idx];
    float nr = fr * fdr - fi * fdi + accR;
    float ni = fr * fdi + fi * fdr + accI;
    s_cat[b * (2 * DIMD) + n]        = nr;
    s_cat[b * (2 * DIMD) + DIMD + n] = ni;
    out_flux[2 * idx]     = nr;   // interleaved complex64
    out_flux[2 * idx + 1] = ni;
}

// ---------------------------------------------------------------------------
// Flux path stage 2: proj = s_cat @ Wp^T + bp -> source; gate = clip(sigmoid)
// ---------------------------------------------------------------------------
__global__ void proj_kernel(const float* __restrict__ s_cat,
                            const float* __restrict__ Wp,
                            const float* __restrict__ bp,
                            const float* __restrict__ Wg,
                            const float* __restrict__ bg,
                            float* __restrict__ src_re,
                            float* __restrict__ src_im,
                            float* __restrict__ gate) {
    int b = blockIdx.x;            // 0..7
    int j = threadIdx.x;           // 0..511
    const float* s  = s_cat + b * (2 * DIMD);
    const float* wp = Wp + (size_t)j * (2 * DIMD);
    float acc = bp[j];
    for (int k = 0; k < 2 * DIMD; ++k) acc = fmaf(s[k], wp[k], acc);
    if (j < DIMD) src_re[b * DIMD + j] = acc;
    else          src_im[b * DIMD + (j - DIMD)] = acc;
    if (j < DIMD) {
        const float* wg = Wg + (size_t)j * (2 * DIMD);
        float ag = bg[j];
        for (int k = 0; k < 2 * DIMD; ++k) ag = fmaf(s[k], wg[k], ag);
        float gv = 1.0f / (1.0f + __expf(-ag));
        gv = fminf(fmaxf(gv, 0.01f), 0.99f);
        gate[b * DIMD + j] = gv;
    }
}

// ---------------------------------------------------------------------------
// Per-channel op_decay = exp(lam*dt) and op_forcing = (op_decay-1)/lam_safe.
// ---------------------------------------------------------------------------
__global__ void ops_kernel(const float* __restrict__ lam_re,
                           const float* __restrict__ lam_im,
                           const int* __restrict__ dt,
                           float* __restrict__ opd_re, float* __restrict__ opd_im,
                           float* __restrict__ opf_re, float* __restrict__ opf_im) {
    int n = threadIdx.x;
    float dtr = (float)dt[0];                    // DT_REF = 1.0
    float lr = fminf(fmaxf(lam_re[n], -0.3f), 0.3f);
    float li = lam_im[n];
    float em  = __expf(lr * dtr);
    float odr = em * __cosf(li * dtr);
    float odi = em * __sinf(li * dtr);
    float sgn = (lr + 1e-12f) >= 0.0f ? 1.0f : -1.0f;
    float lsr = lr + 1e-8f * sgn;
    float lsi = li;
    float nr = odr - 1.0f, ni = odi;
    float den = lsr * lsr + lsi * lsi;
    opd_re[n] = odr; opd_im[n] = odi;
    opf_re[n] = (nr * lsr + ni * lsi) / den;
    opf_im[n] = (ni * lsr - nr * lsi) / den;
}

// ---------------------------------------------------------------------------
// Main kernel. 1024 blocks x 256 threads (8 waves). Each block: 32 rows.
// Each wave: 2 row-tiles x 2 col-tiles of 16x16 complex output
// (16 v_wmma_f32_16x16x4_f32 per K-step of 4). X tile staged via TDM.
// ---------------------------------------------------------------------------
__global__ __launch_bounds__(256) void propagate_kernel(
    const float* __restrict__ x_re, const float* __restrict__ x_im,
    const float* __restrict__ h_re, const float* __restrict__ h_im,
    const float* __restrict__ mt_re, const float* __restrict__ mt_im,
    const float* __restrict__ src_re, const float* __restrict__ src_im,
    const float* __restrict__ gate,
    const float* __restrict__ opd_re, const float* __restrict__ opd_im,
    const float* __restrict__ opf_re, const float* __restrict__ opf_im,
    float* __restrict__ out) {

    // Single static LDS allocation -> LDS offset 0. Pitch 260 floats so the
    // 16 A-rows hit distinct bank groups (260 mod 64 = 4 banks/row).
    __shared__ float Xs[2][TILE_M][XPITCH];       // 66,560 B (<320KB/WGP)

    const int tid = threadIdx.x;
    const int r0  = blockIdx.x * TILE_M;          // first global row of tile

#ifdef USE_TDM
    if (tid < 32) {                               // TDM ignores EXEC: 1 wave
        // D# group1: wg_mask=0 | data_size=4B(2) | pad_enable |
        // pad_interval=256DW(7) | pad_amount=4DW(3); tensor 256x32 = tile;
        // tensor_dim0_stride = 256.
        const i32x8 g1 = { 0x07D20000, 0x01000000, 0x00200000, 0x01000000,
                           32, 256, 0, 0 };
        const i32x4 z4 = { 0, 0, 0, 0 };
        const i32x8 z8 = { 0, 0, 0, 0, 0, 0, 0, 0 };
        uint64_t ga_re = (uint64_t)(uintptr_t)(x_re + (size_t)r0 * DIMD);
        uint64_t ga_im = (uint64_t)(uintptr_t)(x_im + (size_t)r0 * DIMD);
        // D# group0: count=1 | lds_addr | global_addr | type=2("image")
        u32x4 g0_re = { 1u, 0u,
                        (uint32_t)ga_re,
                        ((uint32_t)(ga_re >> 32) & 0x01FFFFFFu) | 0x80000000u };
        u32x4 g0_im = { 1u, (uint32_t)(TILE_M * XPITCH * 4),
                        (uint32_t)ga_im,
                        ((uint32_t)(ga_im >> 32) & 0x01FFFFFFu) | 0x80000000u };
        __builtin_amdgcn_tensor_load_to_lds(g0_re, g1, z4, z4, z8, 0);
        __builtin_amdgcn_tensor_load_to_lds(g0_im, g1, z4, z4, z8, 0);
        __builtin_amdgcn_s_wait_tensorcnt(0);
    }
    __syncthreads();
    // The TDM builtin carries no alias info for Xs: tell the compiler the
    // LDS tile was written, so the A-fragment ds_loads are not folded away.
    asm volatile("" : : "r"(&Xs[0][0][0]) : "memory");
#else
    #pragma unroll 4
    for (int i = 0; i < TILE_M; ++i) {
        Xs[0][i][tid] = x_re[(size_t)(r0 + i) * DIMD + tid];
        Xs[1][i][tid] = x_im[(size_t)(r0 + i) * DIMD + tid];
    }
    __syncthreads();
#endif

    const int wave = tid >> 5;                    // 0..7
    const int lane = tid & 31;
    const int m    = lane & 15;                   // A row / D column-lane
    const int hh   = lane >> 4;                   // lane half
    const int koff = 2 * hh;                      // A/B K sub-offset

    const int ng0 = wave * 16 + m;                // col-tile 0 global column
    const int ng1 = (wave + 8) * 16 + m;          // col-tile 1 global column
    const float* pbre[2] = { mt_re + (size_t)ng0 * DIMD, mt_re + (size_t)ng1 * DIMD };
    const float* pbim[2] = { mt_im + (size_t)ng0 * DIMD, mt_im + (size_t)ng1 * DIMD };

    v8f cre[2][2] = { { (v8f){}, (v8f){} }, { (v8f){}, (v8f){} } };
    v8f cim[2][2] = { { (v8f){}, (v8f){} }, { (v8f){}, (v8f){} } };

    for (int k0 = 0; k0 < DIMD; k0 += 4) {
        const int kb = k0 + koff;
        v2f a_re[2], a_im[2], a_imn[2];
        #pragma unroll
        for (int rt = 0; rt < 2; ++rt) {          // 2 row-tiles: rows m, m+16
            a_re[rt]  = *(const v2f*)&Xs[0][rt * 16 + m][kb];
            a_im[rt]  = *(const v2f*)&Xs[1][rt * 16 + m][kb];
            a_imn[rt] = -a_im[rt];                // f32 WMMA has no A/B negate
        }
        #pragma unroll
        for (int nt = 0; nt < 2; ++nt) {          // 2 col-tiles
            v2f b_re = *(const v2f*)(pbre[nt] + kb);
            v2f b_im = *(const v2f*)(pbim[nt] + kb);
            #pragma unroll
            for (int rt = 0; rt < 2; ++rt) {
                // C_re += A_re*B_re - A_im*B_im ; C_im += A_re*B_im + A_im*B_re
                cre[rt][nt] = __builtin_amdgcn_wmma_f32_16x16x4_f32(
                    false, a_re[rt], false, b_re, (short)0, cre[rt][nt], false, false);
                cre[rt][nt] = __builtin_amdgcn_wmma_f32_16x16x4_f32(
                    false, a_imn[rt], false, b_im, (short)0, cre[rt][nt], false, false);
                cim[rt][nt] = __builtin_amdgcn_wmma_f32_16x16x4_f32(
                    false, a_re[rt], false, b_im, (short)0, cim[rt][nt], false, false);
                cim[rt][nt] = __builtin_amdgcn_wmma_f32_16x16x4_f32(
                    false, a_im[rt], false, b_re, (short)0, cim[rt][nt], false, false);
            }
        }
    }

    // Fused epilogue: h_next = h*op_decay + (x_tilde*g + src*(1-g))*op_forcing
    const int b = blockIdx.x >> 7;                // 128 blocks per batch entry
    const int ngs[2] = { ng0, ng1 };
    #pragma unroll
    for (int nt = 0; nt < 2; ++nt) {
        const int ng = ngs[nt];
        const float g   = gate[b * DIMD + ng];
        const float srw = src_re[b * DIMD + ng] * (1.0f - g);
        const float siw = src_im[b * DIMD + ng] * (1.0f - g);
        const float odr = opd_re[ng], odi = opd_im[ng];
        const float ofr = opf_re[ng], ofi = opf_im[ng];
        #pragma unroll
        for (int rt = 0; rt < 2; ++rt) {
            #pragma unroll
            for (int v = 0; v < 8; ++v) {
                const size_t row = (size_t)(r0 + rt * 16 + v + 8 * hh);
                const float xr = cre[rt][nt][v], xi = cim[rt][nt][v];
                const float fr = fmaf(xr, g, srw);
                const float fi = fmaf(xi, g, siw);
                const float hr = h_re[row * DIMD + ng];
                const float hi = h_im[row * DIMD + ng];
                v2f o;
                o.x = hr * odr - hi * odi + fr * ofr - fi * ofi;
                o.y = hr * odi + hi * odr + fr * ofi + fi * ofr;
                *(v2f*)(out + (row * DIMD + ng) * 2) = o;  // interleaved c64
            }
        }
    }
}

// ---------------------------------------------------------------------------
extern "C" void kernel_launch(void* const* d_in, const int* in_sizes, int n_in,
                              void* d_out, int out_size, void* d_ws, size_t ws_size,
                              hipStream_t stream) {
    (void)in_sizes; (void)n_in; (void)out_size; (void)ws_size;

    const float* h_re     = (const float*)d_in[0];
    const float* h_im     = (const float*)d_in[1];
    const float* x_re     = (const float*)d_in[2];
    const float* x_im     = (const float*)d_in[3];
    const float* xg_re    = (const float*)d_in[4];
    const float* xg_im    = (const float*)d_in[5];
    const float* flux_re  = (const float*)d_in[6];
    const float* flux_im  = (const float*)d_in[7];
    const float* U_re     = (const float*)d_in[8];
    const float* U_im     = (const float*)d_in[9];
    const float* dft_w    = (const float*)d_in[10];
    const float* decay_re = (const float*)d_in[11];
    const float* decay_im = (const float*)d_in[12];
    const float* Wm       = (const float*)d_in[13];
    const float* bm       = (const float*)d_in[14];
    const float* Wp       = (const float*)d_in[15];
    const float* bp       = (const float*)d_in[16];
    const float* Wg       = (const float*)d_in[17];
    const float* bg       = (const float*)d_in[18];
    const float* lam_re   = (const float*)d_in[19];
    const float* lam_im   = (const float*)d_in[20];
    const int*   dt       = (const int*)d_in[21];

    float* out = (float*)d_out;
    float* ws  = (float*)d_ws;

    float* mt_re  = ws + WS_MT_RE;
    float* mt_im  = ws + WS_MT_IM;
    float* s_cat  = ws + WS_SCAT;
    float* src_re = ws + WS_SRC_RE;
    float* src_im = ws + WS_SRC_IM;
    float* gate   = ws + WS_GATE;
    float* opd_re = ws + WS_OPD_RE;
    float* opd_im = ws + WS_OPD_IM;
    float* opf_re = ws + WS_OPF_RE;
    float* opf_im = ws + WS_OPF_IM;

    float* out_flux = out + (size_t)2 * TBROWS * DIMD;  // tuple part 2

    build_m_kernel<<<DIMD, 256, 0, stream>>>(U_re, U_im, dft_w, mt_re, mt_im);
    flux_kernel<<<BATCH, 256, 0, stream>>>(xg_re, xg_im, flux_re, flux_im,
                                           Wm, bm, decay_re, decay_im,
                                           s_cat, out_flux);
    proj_kernel<<<BATCH, 2 * DIMD, 0, stream>>>(s_cat, Wp, bp, Wg, bg,
                                                src_re, src_im, gate);
    ops_kernel<<<1, DIMD, 0, stream>>>(lam_re, lam_im, dt,
                                       opd_re, opd_im, opf_re, opf_im);
    propagate_kernel<<<TBROWS / TILE_M, 256, 0, stream>>>(
        x_re, x_im, h_re, h_im, mt_re, mt_im,
        src_re, src_im, gate, opd_re, opd_im, opf_re, opf_im, out);
}